// DepthLoss3D_47433618817308
// MI455X (gfx1250) — compile-verified
//
#include <hip/hip_runtime.h>
#include <hip/hip_bf16.h>
#include <stdint.h>

// Problem constants (fixed by the reference: h = w = d = 16, N = 4096).
#define NTOT   4096
#define NELM   (NTOT * 3)          // 12288 f32 = 48 KB
#define BLOCK  256                 // 8 wave32s per block
#define WAVES_PER_BLOCK 8
#define NBLOCKS ((NTOT / 16) / WAVES_PER_BLOCK)   // 256 i-tiles / 8 = 32 blocks

typedef __attribute__((ext_vector_type(2))) float v2f;
typedef __attribute__((ext_vector_type(8))) float v8f;

// unravel_index(i, (16,16,16)) -> coordinate for channel c
__device__ __forceinline__ float coordOf(int i, int c) {
    int v = (c == 0) ? (i >> 8) : (c == 1) ? ((i >> 4) & 15) : (i & 15);
    return (float)v;
}

__global__ void DepthLoss3D_zero(float* __restrict__ out) {
    if (threadIdx.x < 3) out[threadIdx.x] = 0.0f;
}

__global__ __launch_bounds__(BLOCK)
void DepthLoss3D_main(const float* __restrict__ pred,
                      const float* __restrict__ spac,
                      float* __restrict__ out) {
    __shared__ float sPred[NELM];   // all 4096x3 predictions, 48 KB
    const int tid = threadIdx.x;

    // ---- Stage predictions into LDS (CDNA5 async global->LDS path if available) ----
#if defined(__gfx1250__) && __has_builtin(__builtin_amdgcn_global_load_async_to_lds_b32)
    for (int k = tid; k < NELM; k += BLOCK) {
        __attribute__((address_space(1))) int* g =
            (__attribute__((address_space(1))) int*)(uintptr_t)(pred + k);
        __attribute__((address_space(3))) int* l =
            (__attribute__((address_space(3))) int*)(uint32_t)(uintptr_t)(&sPred[k]);
        __builtin_amdgcn_global_load_async_to_lds_b32(g, l, 0, 0);
    }
#if __has_builtin(__builtin_amdgcn_s_wait_asynccnt)
    __builtin_amdgcn_s_wait_asynccnt(0);
#else
    asm volatile("s_wait_asynccnt 0" ::: "memory");
#endif
#else
    for (int k = tid; k < NELM; k += BLOCK) sPred[k] = pred[k];
#endif
    __syncthreads();

    const int lane  = tid & 31;
    const int wave  = tid >> 5;
    const int iTile = blockIdx.x * WAVES_PER_BLOCK + wave;  // 0..255
    const int i0    = iTile << 4;
    const int l15   = lane & 15;
    const bool lo   = (lane < 16);
    const int mBase = (lane >> 4) << 3;   // C/D layout: M = r + 8*(lane>=16)

    float acc[3] = {0.0f, 0.0f, 0.0f};

    for (int c = 0; c < 3; ++c) {
        const float sc2 = spac[c] * 2.0f;   // STEP = 2.0
        const int   ig  = i0 + l15;
        const float pi  = sPred[ig * 3 + c];
        const float xi  = coordOf(ig, c) * sc2;

#if defined(__gfx1250__) && __has_builtin(__builtin_amdgcn_wmma_f32_16x16x4_f32)
        // A (16x4): row m = (p_i[m], 1, 0, 0). Lanes 0-15: K=0,1; lanes 16-31: K=2,3 (zeros).
        v2f aP; aP.x = lo ? pi : 0.0f;  aP.y = lo ? 1.0f : 0.0f;
        v2f aX; aX.x = lo ? xi : 0.0f;  aX.y = aP.y;
#endif

        for (int j0 = 0; j0 < NTOT; j0 += 16) {
            v8f dmT, stT;
#if defined(__gfx1250__) && __has_builtin(__builtin_amdgcn_wmma_f32_16x16x4_f32)
            {
                const int   jg = j0 + l15;
                const float pj = sPred[jg * 3 + c];
                const float xj = coordOf(jg, c) * sc2;
                // B (4x16): col n = (1, -p_j[n], 0, 0).
                v2f bP; bP.x = lo ? 1.0f : 0.0f;  bP.y = lo ? -pj : 0.0f;
                v2f bX; bX.x = bP.x;              bX.y = lo ? -xj : 0.0f;
                v8f z = {0.f, 0.f, 0.f, 0.f, 0.f, 0.f, 0.f, 0.f};
                // D[m,n] = p_i[m] - p_j[n]  (rank-2 outer difference via matrix pipe)
                dmT = __builtin_amdgcn_wmma_f32_16x16x4_f32(
                          false, aP, false, bP, (short)0, z, false, false);
                stT = __builtin_amdgcn_wmma_f32_16x16x4_f32(
                          false, aX, false, bX, (short)0, z, false, false);
            }
#else
            {   // Bit-identical scalar fallback matching the C/D VGPR layout.
                const int   ng = j0 + l15;
                const float pj = sPred[ng * 3 + c];
                const float xj = coordOf(ng, c) * sc2;
#pragma unroll
                for (int r = 0; r < 8; ++r) {
                    const int mg = i0 + mBase + r;
                    dmT[r] = sPred[mg * 3 + c] - pj;
                    stT[r] = coordOf(mg, c) * sc2 - xj;
                }
            }
#endif
            // Masked where-chain + |.| reduction (pure VALU on the accumulator tile).
            float a = 0.0f;
#pragma unroll
            for (int r = 0; r < 8; ++r) {
                float v = dmT[r];
                float s = stT[r];
                v = (v >= 0.0f) ? (v - 0.2f * s) : v;
                v = (v >= 0.0f) ? fmaxf(v - 0.8f * s, 0.0f) : v;
                v = (s >= 0.0f) ? v : 0.0f;
                v = (s == 0.0f) ? 0.5f * v : v;
                a += fabsf(v);
            }
            acc[c] += a;
        }
    }

    // ---- Wave reduction (wave32) + scaled atomic accumulation ----
    const float inv = 1.0f / 16777216.0f;   // 1/N^2 = 2^-24, exact
#pragma unroll
    for (int c = 0; c < 3; ++c) {
        float v = acc[c];
        for (int off = 16; off > 0; off >>= 1) v += __shfl_xor(v, off, 32);
        if (lane == 0) atomicAdd(&out[c], v * inv);
    }
}

extern "C" void kernel_launch(void* const* d_in, const int* in_sizes, int n_in,
                              void* d_out, int out_size, void* d_ws, size_t ws_size,
                              hipStream_t stream) {
    const float* pred = (const float*)d_in[0];   // [4096, 3] f32
    const float* spac = (const float*)d_in[1];   // [3, 1]  f32
    float* out = (float*)d_out;                  // [3]     f32

    DepthLoss3D_zero<<<1, 32, 0, stream>>>(out);
    DepthLoss3D_main<<<NBLOCKS, BLOCK, 0, stream>>>(pred, spac, out);
}